// MixedAttention_68685116998000
// MI455X (gfx1250) — compile-verified
//
#include <hip/hip_runtime.h>
#include <math.h>

// ---------------------------------------------------------------------------
// MixedAttention for MI455X (gfx1250, wave32).
// All GEMM-like math runs on V_WMMA_F32_16X16X4_F32 (fp32 in / fp32 acc):
// the whole problem (~38 MB) lives in the 192 MB L2, so it is compute bound
// and fp32 WMMA preserves reference accuracy.
//
// GEMM waves own 32x64 output tiles (8 accumulators): each K=4 step issues
// 2 A-fragment loads + 8 B loads for 8 WMMAs (1.25 loads/WMMA), keeping the
// matrix pipe ahead of the VMEM issue rate.
// ---------------------------------------------------------------------------

typedef __attribute__((ext_vector_type(2))) float v2f;
typedef __attribute__((ext_vector_type(8))) float v8f;

#define DIMC      512
#define HEADS     8
#define HEAD_DIM  32
#define KS        7
#define NHALF     3
#define HW        48
#define NTOK      (HW * HW)          // 2304 tokens per batch
#define BATCH     2
#define TTOT      (BATCH * NTOK)     // 4608 tokens total
#define QKVC      (3 * DIMC)         // 1536
#define RPBW      (2 * KS - 1)       // 13
#define SCALE     0.17677669529663687f   // 1/sqrt(32)

static __device__ __forceinline__ v8f wmma4(v2f a, v2f b, v8f c) {
  // 8-arg form: (neg_a, A, neg_b, B, c_mod, C, reuse_a, reuse_b)
  return __builtin_amdgcn_wmma_f32_16x16x4_f32(false, a, false, b, (short)0, c,
                                               false, false);
}

// ---------------------------------------------------------------------------
// WMMA GEMM:  C[M,N] = A[M,K] @ B[K,N] + bias[N]
// One wave owns one 32x64 tile (2 M-subtiles x 4 N-subtiles).
// M multiple of 32, N multiple of 64, K multiple of 4.
// A-fragment per lane: M = lane%16, K = 2*(lane/16)+{0,1}  (ISA 16x4 layout)
// B-fragment per lane: N = lane%16, K = 2*(lane/16)+{0,1}
// C/D per lane:        N = lane%16, VGPR r -> M = r + 8*(lane/16)
// ---------------------------------------------------------------------------
__global__ void wmma_gemm_bias(const float* __restrict__ A,
                               const float* __restrict__ Bw,
                               const float* __restrict__ bias,
                               float* __restrict__ C,
                               int M, int N, int K) {
  const int lane = threadIdx.x & 31;
  const int wave = threadIdx.x >> 5;
  const int lm   = lane & 15;
  const int hi   = lane >> 4;          // 0: lanes 0-15, 1: lanes 16-31
  const int tilesN = N >> 6;           // 64-wide supertiles
  const int tile = blockIdx.x * (blockDim.x >> 5) + wave;
  if (tile >= (M >> 5) * tilesN) return;   // wave-uniform guard (EXEC stays full)
  const int m0 = (tile / tilesN) << 5;
  const int n0 = (tile % tilesN) << 6;

  const float* arow0 = A + (size_t)(m0 + lm) * K + 2 * hi;
  const float* arow1 = arow0 + (size_t)16 * K;
  const float* brow  = Bw + (size_t)(2 * hi) * N + n0 + lm;

  v8f acc[2][4];
#pragma unroll
  for (int mt = 0; mt < 2; ++mt)
#pragma unroll
    for (int nt = 0; nt < 4; ++nt)
      acc[mt][nt] = (v8f){0.f,0.f,0.f,0.f,0.f,0.f,0.f,0.f};

#pragma unroll 2
  for (int k = 0; k < K; k += 4) {
    const v2f a0 = *(const v2f*)(arow0 + k);   // K, K+1 contiguous
    const v2f a1 = *(const v2f*)(arow1 + k);
    const float* br0 = brow + (size_t)k * N;
    const float* br1 = br0 + N;
#pragma unroll
    for (int nt = 0; nt < 4; ++nt) {
      v2f b;
      b.x = br0[nt * 16];
      b.y = br1[nt * 16];
      acc[0][nt] = wmma4(a0, b, acc[0][nt]);
      acc[1][nt] = wmma4(a1, b, acc[1][nt]);
    }
  }

#pragma unroll
  for (int nt = 0; nt < 4; ++nt) {
    const float bv = bias[n0 + nt * 16 + lm];
#pragma unroll
    for (int mt = 0; mt < 2; ++mt) {
#pragma unroll
      for (int r = 0; r < 8; ++r) {
        C[(size_t)(m0 + mt * 16 + r + 8 * hi) * N + n0 + nt * 16 + lm] =
            acc[mt][nt][r] + bv;
      }
    }
  }
}

// ---------------------------------------------------------------------------
// Neighborhood attention (7x7 window). One wave per (b, head, i, j),
// lane = head dim. Coalesced k/v reads, wave-shuffle reductions.
// qkv layout per token: [3*DIM]; NA half = [0:768) as [3][8][32].
// Output goes to attn[token, 0:256).
// ---------------------------------------------------------------------------
__global__ void na_attention_kernel(const float* __restrict__ qkv,
                                    const float* __restrict__ rpb,
                                    float* __restrict__ attn) {
  const int lane = threadIdx.x & 31;
  const int gid  = blockIdx.x * (blockDim.x >> 5) + (threadIdx.x >> 5);
  const int j = gid % HW;
  const int i = (gid / HW) % HW;
  const int h = (gid / (HW * HW)) % HEADS;
  const int b = gid / (HW * HW * HEADS);

  const int ni = min(max(i - NHALF, 0), HW - KS);
  const int nj = min(max(j - NHALF, 0), HW - KS);
  const int pi = NHALF + max(NHALF - i, 0) + ((i + NHALF >= HW) ? (HW - i - 1 - NHALF) : 0);
  const int pj = NHALF + max(NHALF - j, 0) + ((j + NHALF >= HW) ? (HW - j - 1 - NHALF) : 0);

  const int t = (b * HW + i) * HW + j;
  const float qd = qkv[(size_t)t * QKVC + h * HEAD_DIM + lane] * SCALE;

  float s0 = -INFINITY, s1 = -INFINITY;   // lane l holds scores for nbr l, l+32
  for (int n = 0; n < KS * KS; ++n) {
    const int ki = n / KS, kj = n % KS;
    const int tk = (b * HW + ni + ki) * HW + nj + kj;
    float p = qd * qkv[(size_t)tk * QKVC + 256 + h * HEAD_DIM + lane];
#pragma unroll
    for (int off = 16; off >= 1; off >>= 1) p += __shfl_xor(p, off, 32);
    p += rpb[(h * RPBW + pi + ki) * RPBW + pj + kj];
    if (n < 32) { if (lane == n)      s0 = p; }
    else        { if (lane == n - 32) s1 = p; }
  }
  float pm = fmaxf(s0, s1);
#pragma unroll
  for (int off = 16; off >= 1; off >>= 1) pm = fmaxf(pm, __shfl_xor(pm, off, 32));
  float e0 = __expf(s0 - pm);
  float e1 = (lane < KS * KS - 32) ? __expf(s1 - pm) : 0.f;
  float z = e0 + e1;
#pragma unroll
  for (int off = 16; off >= 1; off >>= 1) z += __shfl_xor(z, off, 32);

  float o = 0.f;
  for (int n = 0; n < KS * KS; ++n) {
    const int ki = n / KS, kj = n % KS;
    const int tk = (b * HW + ni + ki) * HW + nj + kj;
    float src = (n < 32) ? e0 : e1;       // uniform select
    float a = __shfl(src, n & 31, 32);
    o += a * qkv[(size_t)tk * QKVC + 512 + h * HEAD_DIM + lane];
  }
  attn[(size_t)t * DIMC + h * HEAD_DIM + lane] = o / z;
}

// ---------------------------------------------------------------------------
// Full self-attention, flash style. One wave per 16-query tile per (b,h).
// S = Q.K^T via 8x wmma_f32_16x16x4, online softmax with 16-lane xor-shuffle
// row reductions, P transposed C-layout -> A-layout via a wave-private
// padded LDS tile, then P.V via 8x wmma. Self half of qkv = [768:1536).
// Output goes to attn[token, 256:512).
// ---------------------------------------------------------------------------
__global__ void self_attention_kernel(const float* __restrict__ qkv,
                                      float* __restrict__ attn) {
  __shared__ float pls[8][16 * 17];      // 8 waves/block, padded 16x16 P tile
  const int lane = threadIdx.x & 31;
  const int wave = threadIdx.x >> 5;
  const int lm   = lane & 15;
  const int hi   = lane >> 4;
  const int gid  = blockIdx.x * (blockDim.x >> 5) + wave;
  const int qt = gid % (NTOK / 16);
  const int h  = (gid / (NTOK / 16)) % HEADS;
  const int b  = gid / ((NTOK / 16) * HEADS);
  const int q0 = qt * 16;

  const size_t tb = (size_t)b * NTOK;
  auto QOFF = [&](int n, int d) -> size_t {
    return (tb + n) * QKVC + 768 + (size_t)h * HEAD_DIM + d;   // q; k:+256 v:+512
  };

  // Q fragments for the whole d=32 reduction (8 K-steps of 4), scale folded in.
  v2f qa[8];
#pragma unroll
  for (int kc = 0; kc < 8; ++kc) {
    const int d = kc * 4 + 2 * hi;
    v2f q = *(const v2f*)(qkv + QOFF(q0 + lm, d));
    qa[kc].x = q.x * SCALE;
    qa[kc].y = q.y * SCALE;
  }

  float m8[8], l8[8];
#pragma unroll
  for (int r = 0; r < 8; ++r) { m8[r] = -INFINITY; l8[r] = 0.f; }
  v8f acc0 = {0.f,0.f,0.f,0.f,0.f,0.f,0.f,0.f};
  v8f acc1 = {0.f,0.f,0.f,0.f,0.f,0.f,0.f,0.f};

  float* myP = &pls[wave][0];

  for (int kt = 0; kt < NTOK / 16; ++kt) {
    const int k0 = kt * 16;
    if (kt + 1 < NTOK / 16)   // uniform; emits global_prefetch_b8
      __builtin_prefetch(qkv + QOFF(k0 + 16 + lm, 2 * hi) + 256, 0, 0);

    // S = Q . K^T  (16x16)
    v8f S = {0.f,0.f,0.f,0.f,0.f,0.f,0.f,0.f};
#pragma unroll
    for (int kc = 0; kc < 8; ++kc) {
      const int d = kc * 4 + 2 * hi;
      v2f kb = *(const v2f*)(qkv + QOFF(k0 + lm, d) + 256);
      S = wmma4(qa[kc], kb, S);
    }

    // Online softmax per row; stash exp(S) into LDS in [query][key] order.
#pragma unroll
    for (int r = 0; r < 8; ++r) {
      float t = S[r];
#pragma unroll
      for (int off = 8; off >= 1; off >>= 1) t = fmaxf(t, __shfl_xor(t, off, 32));
      const float nm = fmaxf(m8[r], t);
      const float p  = __expf(S[r] - nm);
      float rs = p;
#pragma unroll
      for (int off = 8; off >= 1; off >>= 1) rs += __shfl_xor(rs, off, 32);
      const float sc = __expf(m8[r] - nm);
      l8[r] = l8[r] * sc + rs;
      m8[r] = nm;
      acc0[r] *= sc;
      acc1[r] *= sc;
      myP[(r + 8 * hi) * 17 + lm] = p;    // wave-private; DS ops in-order
    }

    // acc += P . V   (P read back from LDS in A-layout)
#pragma unroll
    for (int kc = 0; kc < 4; ++kc) {
      const int dk = kc * 4 + 2 * hi;     // local key index
      v2f pa = *(const v2f*)(myP + lm * 17 + dk);
#pragma unroll
      for (int dh = 0; dh < 2; ++dh) {
        const int d = dh * 16 + lm;
        v2f vb;
        vb.x = qkv[QOFF(k0 + dk,     d) + 512];
        vb.y = qkv[QOFF(k0 + dk + 1, d) + 512];
        if (dh == 0) acc0 = wmma4(pa, vb, acc0);
        else         acc1 = wmma4(pa, vb, acc1);
      }
    }
  }

#pragma unroll
  for (int r = 0; r < 8; ++r) {
    const float inv = 1.0f / l8[r];
    const size_t ob = (tb + q0 + r + 8 * hi) * DIMC + 256 + (size_t)h * HEAD_DIM;
    attn[ob + lm]      = acc0[r] * inv;
    attn[ob + 16 + lm] = acc1[r] * inv;
  }
}

// ---------------------------------------------------------------------------
extern "C" void kernel_launch(void* const* d_in, const int* in_sizes, int n_in,
                              void* d_out, int out_size, void* d_ws, size_t ws_size,
                              hipStream_t stream) {
  const float* x      = (const float*)d_in[0];
  const float* W_qkv  = (const float*)d_in[1];
  const float* b_qkv  = (const float*)d_in[2];
  const float* rpb    = (const float*)d_in[3];
  const float* W_proj = (const float*)d_in[4];
  const float* b_proj = (const float*)d_in[5];
  float* out = (float*)d_out;

  float* qkv  = (float*)d_ws;                       // [4608, 1536]  28.3 MB
  float* attn = qkv + (size_t)TTOT * QKVC;          // [4608,  512]   9.4 MB

  // 1) qkv = x @ W_qkv + b_qkv        (32x64 tiles: 144*24=3456, 8 waves/blk)
  {
    const int tiles = (TTOT / 32) * (QKVC / 64);
    wmma_gemm_bias<<<tiles / 8, 256, 0, stream>>>(x, W_qkv, b_qkv, qkv,
                                                  TTOT, QKVC, DIMC);
  }
  // 2) neighborhood attention -> attn[:, 0:256)
  na_attention_kernel<<<(BATCH * HEADS * HW * HW) / 8, 256, 0, stream>>>(
      qkv, rpb, attn);
  // 3) full self-attention -> attn[:, 256:512)
  self_attention_kernel<<<(BATCH * HEADS * (NTOK / 16)) / 8, 256, 0, stream>>>(
      qkv, attn);
  // 4) out = attn @ W_proj + b_proj   (32x64 tiles: 144*8=1152)
  {
    const int tiles = (TTOT / 32) * (DIMC / 64);
    wmma_gemm_bias<<<tiles / 8, 256, 0, stream>>>(attn, W_proj, b_proj, out,
                                                  TTOT, DIMC, DIMC);
  }
}